// Canny_30897994727999
// MI455X (gfx1250) — compile-verified
//
#include <hip/hip_runtime.h>
#include <math.h>

#ifndef __has_builtin
#define __has_builtin(x) 0
#endif

#define TILE 32
#define IMGS 40           // TILE + 8  (sobel +-2, NMS +-1, hyst +-1)
#define GMS  36           // TILE + 4
#define GAS  34           // TILE + 2

// gcc-style vector type matching the builtin's expected 'V4i *' parameter
typedef int v4i_g __attribute__((vector_size(16)));

// ---------------------------------------------------------------------------
// gfx1250 async global->LDS helpers (CDNA5 ASYNCcnt path), with fallbacks
// ---------------------------------------------------------------------------
__device__ __forceinline__ void async_ld_b32(const float* g, float* l) {
#if defined(__AMDGCN__) && __has_builtin(__builtin_amdgcn_global_load_async_to_lds_b32)
  __builtin_amdgcn_global_load_async_to_lds_b32((int*)g, (int*)l, 0, 0);
#else
  *l = *g;
#endif
}

__device__ __forceinline__ void async_ld_b128(const float* g, float* l) {
#if defined(__AMDGCN__) && __has_builtin(__builtin_amdgcn_global_load_async_to_lds_b128)
  __builtin_amdgcn_global_load_async_to_lds_b128((v4i_g*)g, (v4i_g*)l, 0, 0);
#else
  *(float4*)l = *(const float4*)g;
#endif
}

__device__ __forceinline__ void wait_async_all() {
#if defined(__AMDGCN__)
#if __has_builtin(__builtin_amdgcn_s_wait_asynccnt)
  __builtin_amdgcn_s_wait_asynccnt(0);
#else
  asm volatile("s_wait_asynccnt 0" ::: "memory");
#endif
#endif
}

__device__ __forceinline__ int reflect512(int i) {
  if (i < 0) i = -i;
  if (i > 511) i = 1022 - i;
  return i;
}

// ---------------------------------------------------------------------------
// Kernel 0: zero the rescale flag
// ---------------------------------------------------------------------------
__global__ void init_flag(int* flag) {
  if (threadIdx.x == 0 && blockIdx.x == 0) *flag = 0;
}

// ---------------------------------------------------------------------------
// Kernel 1: flag = any(images < -0.001)   (equivalent to min() < -0.001)
// ---------------------------------------------------------------------------
__global__ __launch_bounds__(256) void scan_neg(const float4* __restrict__ x,
                                                int n4, int* __restrict__ flag) {
  int pred = 0;
  for (int i = blockIdx.x * blockDim.x + threadIdx.x; i < n4;
       i += gridDim.x * blockDim.x) {
    float4 v = x[i];
    pred |= (v.x < -0.001f) | (v.y < -0.001f) | (v.z < -0.001f) | (v.w < -0.001f);
  }
  if (__syncthreads_or(pred) && threadIdx.x == 0) atomicOr(flag, 1);
}

// ---------------------------------------------------------------------------
// Kernel 2: fully fused Canny. One block = one 32x32 output tile.
// ---------------------------------------------------------------------------
__global__ __launch_bounds__(256) void canny_fused(
    const float* __restrict__ img,   // [16,3,512,512]
    const float* __restrict__ wxp,   // [25] sobel_x
    const float* __restrict__ wyp,   // [25] sobel_y
    const float* __restrict__ hwp,   // [9]  hysteresis weights (ones)
    const int*   __restrict__ flag,
    float*       __restrict__ out)   // [16,1,512,512]
{
  __shared__ __align__(16) float sC[3][IMGS * IMGS];  // channels; sC[0] reused as lum
  __shared__ float sGM[GMS * GMS];                    // raw gradient magnitude
  __shared__ float sGA[GAS * GAS];                    // post-NMS, post-low-threshold
  __shared__ unsigned char sPH[GMS * GMS];            // phase bucket
  __shared__ float sWX[25], sWY[25], sHW[9];

  const int tid = threadIdx.x;
  const int bx = blockIdx.x, by = blockIdx.y, bz = blockIdx.z;
  const int x0 = bx * TILE, y0 = by * TILE;

  if (tid < 25) { sWX[tid] = wxp[tid]; sWY[tid] = wyp[tid]; }
  else if (tid < 34) { sHW[tid - 25] = hwp[tid - 25]; }

  const float* base = img + (size_t)bz * (3u * 512u * 512u);

  // ---- Stage 1: async-stage the 40x40 tile of all 3 channels into LDS ----
  const bool interior = (bx > 0) & (bx < 15) & (by > 0) & (by < 15);
  if (interior) {
    // 40 rows x 10 float4 per channel, 16B aligned (x0-4 is a multiple of 4)
    for (int i = tid; i < 3 * IMGS * 10; i += 256) {
      int c = i / (IMGS * 10);
      int r = i % (IMGS * 10);
      int ly = r / 10, lx = (r % 10) * 4;
      const float* g = base + (size_t)c * 262144u +
                       (size_t)(y0 - 4 + ly) * 512u + (size_t)(x0 - 4 + lx);
      async_ld_b128(g, &sC[c][ly * IMGS + lx]);
    }
  } else {
    for (int i = tid; i < 3 * IMGS * IMGS; i += 256) {
      int c = i / (IMGS * IMGS);
      int r = i % (IMGS * IMGS);
      int ly = r / IMGS, lx = r % IMGS;
      int gy = reflect512(y0 - 4 + ly);
      int gx = reflect512(x0 - 4 + lx);
      async_ld_b32(base + (size_t)c * 262144u + (size_t)gy * 512u + (size_t)gx,
                   &sC[c][r]);
    }
  }
  wait_async_all();
  __syncthreads();

  // ---- Stage 2: optional rescale + channel L2 norm -> sC[0] (in place) ----
  const bool rs = (flag[0] != 0);
  for (int i = tid; i < IMGS * IMGS; i += 256) {
    float a = sC[0][i], b = sC[1][i], c = sC[2][i];
    if (rs) { a = (a + 1.f) * 0.5f; b = (b + 1.f) * 0.5f; c = (c + 1.f) * 0.5f; }
    sC[0][i] = sqrtf(a * a + b * b + c * c);
  }
  __syncthreads();

  // ---- Stage 3: 5x5 sobel (reflect pad via staged halo), |grad|, phase ----
  for (int i = tid; i < GMS * GMS; i += 256) {
    int ly = i / GMS, lx = i % GMS;
    float sx = 0.f, sy = 0.f;
#pragma unroll
    for (int ky = 0; ky < 5; ky++) {
#pragma unroll
      for (int kx = 0; kx < 5; kx++) {
        float v = sC[0][(ly + ky) * IMGS + (lx + kx)];
        sx = fmaf(sWX[ky * 5 + kx], v, sx);
        sy = fmaf(sWY[ky * 5 + kx], v, sy);
      }
    }
    sGM[i] = sqrtf(sx * sx + sy * sy);
    // phase = mod(round(atan2(sx, sy+1e-5)/(pi/4)) + 4, 8); rintf = round-half-even
    int ph = ((int)rintf(atan2f(sx, sy + 1e-5f) * 1.27323954473516f) + 4) & 7;
    sPH[i] = (unsigned char)ph;
  }
  __syncthreads();

  // ---- Stage 4: NMS (one-hot selection convs == shifted reads, zero-pad
  //               at the true image border) + low threshold ----
  {
    const int offy[8] = {-1, 1, 0, 0, -1, 1, -1, 1};
    const int offx[8] = { 0, 0,-1, 1, -1, 1,  1,-1};
    const int basep[4] = {0, 4, 2, 6};  // SEL_IDS pair base per (phase & 3)
    for (int i = tid; i < GAS * GAS; i += 256) {
      int ly = i / GAS, lx = i % GAS;
      int gmi = (ly + 1) * GMS + (lx + 1);
      float g = sGM[gmi];
      int a = basep[sPH[gmi] & 3];
      int qgx = x0 - 1 + lx, qgy = y0 - 1 + ly;
      float nb0 = 0.f, nb1 = 0.f;
      {
        int dx = offx[a], dy = offy[a];
        if ((unsigned)(qgx + dx) < 512u && (unsigned)(qgy + dy) < 512u)
          nb0 = sGM[gmi + dy * GMS + dx];
      }
      {
        int dx = offx[a + 1], dy = offy[a + 1];
        if ((unsigned)(qgx + dx) < 512u && (unsigned)(qgy + dy) < 512u)
          nb1 = sGM[gmi + dy * GMS + dx];
      }
      float r = ((g < nb0) || (g < nb1)) ? 0.f : g;
      sGA[i] = (r < 0.05f) ? 0.f : r;
    }
  }
  __syncthreads();

  // ---- Stage 5: hysteresis (3x3 reflect-padded conv of high mask) + store ----
  for (int i = tid; i < TILE * TILE; i += 256) {
    int ly = i >> 5, lx = i & 31;
    float g = sGA[(ly + 1) * GAS + (lx + 1)];
    int gx = x0 + lx, gy = y0 + ly;
    float s = 0.f;
#pragma unroll
    for (int dy = -1; dy <= 1; dy++) {
#pragma unroll
      for (int dx = -1; dx <= 1; dx++) {
        int ny = reflect512(gy + dy);
        int nx = reflect512(gx + dx);
        int ay = ny - (y0 - 1), ax = nx - (x0 - 1);   // always within GAS region
        float h = (sGA[ay * GAS + ax] > 0.2f) ? 1.f : 0.f;
        s = fmaf(sHW[(dy + 1) * 3 + (dx + 1)], h, s);
      }
    }
    bool high = g > 0.2f;
    bool weak = (g < 0.2f) && (g > 0.05f);
    float o = high ? g : ((weak && s > 0.f) ? g : 0.f);
    out[((size_t)bz * 512u + (size_t)gy) * 512u + (size_t)gx] = o;
  }
}

// ---------------------------------------------------------------------------
extern "C" void kernel_launch(void* const* d_in, const int* in_sizes, int n_in,
                              void* d_out, int out_size, void* d_ws, size_t ws_size,
                              hipStream_t stream) {
  (void)in_sizes; (void)n_in; (void)out_size; (void)ws_size;
  const float* img = (const float*)d_in[0];   // [16,3,512,512]
  const float* wx  = (const float*)d_in[1];   // [1,1,5,5]
  const float* wy  = (const float*)d_in[2];   // [1,1,5,5]
  /* d_in[3] = sel_w: one-hot 3x3 kernels, folded into NMS shifted reads */
  const float* hw  = (const float*)d_in[4];   // [1,1,3,3]
  float* out = (float*)d_out;
  int* flag = (int*)d_ws;

  init_flag<<<1, 1, 0, stream>>>(flag);

  const int n4 = (16 * 3 * 512 * 512) / 4;
  scan_neg<<<1024, 256, 0, stream>>>((const float4*)img, n4, flag);

  dim3 grid(16, 16, 16);  // 16x16 tiles of 32x32, 16 batch images
  canny_fused<<<grid, 256, 0, stream>>>(img, wx, wy, hw, flag, out);
}